// TinyGAT_7413113553678
// MI455X (gfx1250) — compile-verified
//
#include <hip/hip_runtime.h>
#include <hip/hip_bf16.h>
#include <math.h>

#define NEG_SLOPE 0.2f

typedef float v2f __attribute__((ext_vector_type(2)));
typedef float v8f __attribute__((ext_vector_type(8)));

// ---------- helpers: monotone float<->uint key for atomic float max ----------
__device__ __forceinline__ unsigned fkey(float f) {
    unsigned u = __float_as_uint(f);
    return (u & 0x80000000u) ? ~u : (u | 0x80000000u);
}
__device__ __forceinline__ float funkey(unsigned k) {
    return (k & 0x80000000u) ? __uint_as_float(k & 0x7fffffffu)
                             : __uint_as_float(~k);
}
__device__ __forceinline__ float lrelu(float e) {
    return e >= 0.0f ? e : NEG_SLOPE * e;
}

// ---------- init: zero accumulators, min-keys, seed out with b2 ----------
__global__ void tg_init(float* acc1, unsigned* m1, float* z1,
                        unsigned* m2, float* z2,
                        float* out, const float* __restrict__ b2, int N) {
    int i = blockIdx.x * blockDim.x + threadIdx.x;
    int total = N * 64;
    if (i >= total) return;
    acc1[i] = 0.0f;
    if (i < N * 2) { m1[i] = 0u; z1[i] = 0.0f; }
    if (i < N)     { m2[i] = 0u; z2[i] = 0.0f; }
    if (i < N * 32) out[i] = b2[i & 31];
}

// ---------- layer-1 GEMM: h1[N,64] = x[N,64] @ W1[64,64], WMMA f32 ----------
__global__ __launch_bounds__(128) void tg_gemm1(const float* __restrict__ x,
                                                const float* __restrict__ W1,
                                                float* __restrict__ h1, int N) {
    __shared__ float w[64 * 64];
    int t = threadIdx.x;
    for (int i = t; i < 64 * 64; i += 128) w[i] = W1[i];
    __syncthreads();

    int wave = t >> 5;
    int lane = t & 31;
    int m0 = blockIdx.x * 16;          // node tile
    int n0 = wave * 16;                // output-column tile (4 tiles of 16)
    int lrow  = lane & 15;
    int khalf = (lane >> 4) * 2;       // lanes 0-15 -> K 0,1 ; lanes 16-31 -> K 2,3

    int arow = m0 + lrow; if (arow >= N) arow = N - 1;   // clamp (N%16==0 anyway)

    v8f c = {};
    #pragma unroll
    for (int k0 = 0; k0 < 64; k0 += 4) {
        v2f a, b;
        const float* xr = x + (size_t)arow * 64 + k0 + khalf;
        a.x = xr[0];
        a.y = xr[1];
        b.x = w[(k0 + khalf)     * 64 + n0 + lrow];
        b.y = w[(k0 + khalf + 1) * 64 + n0 + lrow];
        c = __builtin_amdgcn_wmma_f32_16x16x4_f32(false, a, false, b,
                                                  (short)0, c, false, false);
    }
    // C/D layout: VGPR i -> M = i (lanes 0-15) or 8+i (lanes 16-31); N = lane&15
    int mbase = m0 + ((lane >> 4) ? 8 : 0);
    int colg  = n0 + lrow;
    #pragma unroll
    for (int i = 0; i < 8; i++) {
        int row = mbase + i;
        if (row < N) h1[(size_t)row * 64 + colg] = c[i];
    }
}

// ---------- layer-1 per-node attention halves ----------
__global__ void tg_alpha1(const float* __restrict__ h1,
                          const float* __restrict__ a_src,
                          const float* __restrict__ a_dst,
                          float* __restrict__ as1, float* __restrict__ ad1, int N) {
    int n = blockIdx.x * blockDim.x + threadIdx.x;
    if (n >= N) return;
    const float* hr = h1 + (size_t)n * 64;
    #pragma unroll
    for (int h = 0; h < 2; h++) {
        float s = 0.0f, d = 0.0f;
        #pragma unroll
        for (int k = 0; k < 32; k++) {
            float v = hr[h * 32 + k];
            s += v * a_src[h * 32 + k];
            d += v * a_dst[h * 32 + k];
        }
        as1[n * 2 + h] = s;
        ad1[n * 2 + h] = d;
    }
}

// ---------- layer-1 edge passes ----------
__global__ void tg_edge_max1(const long long* __restrict__ ei, long long E, long long ET,
                             const float* __restrict__ as1, const float* __restrict__ ad1,
                             unsigned* m1) {
    long long e = (long long)blockIdx.x * blockDim.x + threadIdx.x;
    if (e >= ET) return;
    long long s, d;
    if (e < E) { s = ei[e]; d = ei[E + e]; } else { s = d = e - E; }
    #pragma unroll
    for (int h = 0; h < 2; h++) {
        float lg = lrelu(as1[s * 2 + h] + ad1[d * 2 + h]);
        atomicMax(&m1[d * 2 + h], fkey(lg));
    }
}

__global__ void tg_edge_sum1(const long long* __restrict__ ei, long long E, long long ET,
                             const float* __restrict__ as1, const float* __restrict__ ad1,
                             const unsigned* __restrict__ m1, float* z1) {
    long long e = (long long)blockIdx.x * blockDim.x + threadIdx.x;
    if (e >= ET) return;
    long long s, d;
    if (e < E) { s = ei[e]; d = ei[E + e]; } else { s = d = e - E; }
    #pragma unroll
    for (int h = 0; h < 2; h++) {
        float lg = lrelu(as1[s * 2 + h] + ad1[d * 2 + h]);
        atomicAdd(&z1[d * 2 + h], __expf(lg - funkey(m1[d * 2 + h])));
    }
}

__global__ void tg_edge_scat1(const long long* __restrict__ ei, long long E, long long ET,
                              const float* __restrict__ as1, const float* __restrict__ ad1,
                              const unsigned* __restrict__ m1, const float* __restrict__ z1,
                              const float* __restrict__ h1, float* acc1) {
    long long tid = (long long)blockIdx.x * blockDim.x + threadIdx.x;
    if (tid >= ET * 2) return;
    long long e = tid >> 1;
    int h = (int)(tid & 1);
    long long s, d;
    if (e < E) { s = ei[e]; d = ei[E + e]; } else { s = d = e - E; }
    float lg = lrelu(as1[s * 2 + h] + ad1[d * 2 + h]);
    float alpha = __expf(lg - funkey(m1[d * 2 + h])) / z1[d * 2 + h];
    const float* src = h1 + (size_t)s * 64 + h * 32;
    float* dst = acc1 + (size_t)d * 64 + h * 32;
    #pragma unroll
    for (int k = 0; k < 32; k++) atomicAdd(&dst[k], alpha * src[k]);
}

// ---------- layer-2 GEMM: h2[N,32] = elu(acc1+b1)[N,64] @ W2[64,32] ----------
__global__ __launch_bounds__(64) void tg_gemm2(const float* __restrict__ acc1,
                                               const float* __restrict__ b1,
                                               const float* __restrict__ W2,
                                               float* __restrict__ h2, int N) {
    __shared__ float w[64 * 32];
    __shared__ float bb[64];
    int t = threadIdx.x;
    for (int i = t; i < 64 * 32; i += 64) w[i] = W2[i];
    if (t < 64) bb[t] = b1[t];
    __syncthreads();

    int wave = t >> 5;
    int lane = t & 31;
    int m0 = blockIdx.x * 16;
    int n0 = wave * 16;                // 2 column tiles of 16
    int lrow  = lane & 15;
    int khalf = (lane >> 4) * 2;

    int arow = m0 + lrow; if (arow >= N) arow = N - 1;

    v8f c = {};
    #pragma unroll
    for (int k0 = 0; k0 < 64; k0 += 4) {
        v2f a, b;
        const float* ar = acc1 + (size_t)arow * 64 + k0 + khalf;
        float v0 = ar[0] + bb[k0 + khalf];
        float v1 = ar[1] + bb[k0 + khalf + 1];
        a.x = v0 > 0.0f ? v0 : __expf(v0) - 1.0f;   // fused ELU
        a.y = v1 > 0.0f ? v1 : __expf(v1) - 1.0f;
        b.x = w[(k0 + khalf)     * 32 + n0 + lrow];
        b.y = w[(k0 + khalf + 1) * 32 + n0 + lrow];
        c = __builtin_amdgcn_wmma_f32_16x16x4_f32(false, a, false, b,
                                                  (short)0, c, false, false);
    }
    int mbase = m0 + ((lane >> 4) ? 8 : 0);
    int colg  = n0 + lrow;
    #pragma unroll
    for (int i = 0; i < 8; i++) {
        int row = mbase + i;
        if (row < N) h2[(size_t)row * 32 + colg] = c[i];
    }
}

// ---------- layer-2 per-node attention halves (1 head, 32 dims) ----------
__global__ void tg_alpha2(const float* __restrict__ h2,
                          const float* __restrict__ a_src,
                          const float* __restrict__ a_dst,
                          float* __restrict__ as2, float* __restrict__ ad2, int N) {
    int n = blockIdx.x * blockDim.x + threadIdx.x;
    if (n >= N) return;
    const float* hr = h2 + (size_t)n * 32;
    float s = 0.0f, d = 0.0f;
    #pragma unroll
    for (int k = 0; k < 32; k++) {
        float v = hr[k];
        s += v * a_src[k];
        d += v * a_dst[k];
    }
    as2[n] = s;
    ad2[n] = d;
}

// ---------- layer-2 edge passes ----------
__global__ void tg_edge_max2(const long long* __restrict__ ei, long long E, long long ET,
                             const float* __restrict__ as2, const float* __restrict__ ad2,
                             unsigned* m2) {
    long long e = (long long)blockIdx.x * blockDim.x + threadIdx.x;
    if (e >= ET) return;
    long long s, d;
    if (e < E) { s = ei[e]; d = ei[E + e]; } else { s = d = e - E; }
    atomicMax(&m2[d], fkey(lrelu(as2[s] + ad2[d])));
}

__global__ void tg_edge_sum2(const long long* __restrict__ ei, long long E, long long ET,
                             const float* __restrict__ as2, const float* __restrict__ ad2,
                             const unsigned* __restrict__ m2, float* z2) {
    long long e = (long long)blockIdx.x * blockDim.x + threadIdx.x;
    if (e >= ET) return;
    long long s, d;
    if (e < E) { s = ei[e]; d = ei[E + e]; } else { s = d = e - E; }
    float lg = lrelu(as2[s] + ad2[d]);
    atomicAdd(&z2[d], __expf(lg - funkey(m2[d])));
}

__global__ void tg_edge_scat2(const long long* __restrict__ ei, long long E, long long ET,
                              const float* __restrict__ as2, const float* __restrict__ ad2,
                              const unsigned* __restrict__ m2, const float* __restrict__ z2,
                              const float* __restrict__ h2, float* out) {
    long long e = (long long)blockIdx.x * blockDim.x + threadIdx.x;
    if (e >= ET) return;
    long long s, d;
    if (e < E) { s = ei[e]; d = ei[E + e]; } else { s = d = e - E; }
    float lg = lrelu(as2[s] + ad2[d]);
    float alpha = __expf(lg - funkey(m2[d])) / z2[d];
    const float* src = h2 + (size_t)s * 32;
    float* dst = out + (size_t)d * 32;
    #pragma unroll
    for (int k = 0; k < 32; k++) atomicAdd(&dst[k], alpha * src[k]);
}

extern "C" void kernel_launch(void* const* d_in, const int* in_sizes, int n_in,
                              void* d_out, int out_size, void* d_ws, size_t ws_size,
                              hipStream_t stream) {
    const float*     x      = (const float*)d_in[0];
    const long long* ei     = (const long long*)d_in[1];
    const float*     W1     = (const float*)d_in[2];
    const float*     a1_src = (const float*)d_in[3];
    const float*     a1_dst = (const float*)d_in[4];
    const float*     b1     = (const float*)d_in[5];
    const float*     W2     = (const float*)d_in[6];
    const float*     a2_src = (const float*)d_in[7];
    const float*     a2_dst = (const float*)d_in[8];
    const float*     b2     = (const float*)d_in[9];
    float* out = (float*)d_out;

    const int       N  = in_sizes[0] / 64;
    const long long E  = (long long)in_sizes[1] / 2;
    const long long ET = E + N;

    // workspace carve (floats)
    float* wsf = (float*)d_ws;
    size_t o = 0;
    float*    h1   = wsf + o; o += (size_t)N * 64;
    float*    acc1 = wsf + o; o += (size_t)N * 64;
    float*    h2   = wsf + o; o += (size_t)N * 32;
    float*    as1  = wsf + o; o += (size_t)N * 2;
    float*    ad1  = wsf + o; o += (size_t)N * 2;
    float*    z1   = wsf + o; o += (size_t)N * 2;
    float*    as2  = wsf + o; o += (size_t)N;
    float*    ad2  = wsf + o; o += (size_t)N;
    float*    z2   = wsf + o; o += (size_t)N;
    unsigned* m1   = (unsigned*)(wsf + o); o += (size_t)N * 2;
    unsigned* m2   = (unsigned*)(wsf + o); o += (size_t)N;

    const int TB = 256;
    const int mtiles = (N + 15) / 16;
    const int eBlocks  = (int)((ET + TB - 1) / TB);
    const int e2Blocks = (int)((ET * 2 + TB - 1) / TB);

    tg_init<<<(N * 64 + TB - 1) / TB, TB, 0, stream>>>(acc1, m1, z1, m2, z2, out, b2, N);

    // layer 1
    tg_gemm1<<<mtiles, 128, 0, stream>>>(x, W1, h1, N);
    tg_alpha1<<<(N + TB - 1) / TB, TB, 0, stream>>>(h1, a1_src, a1_dst, as1, ad1, N);
    tg_edge_max1<<<eBlocks, TB, 0, stream>>>(ei, E, ET, as1, ad1, m1);
    tg_edge_sum1<<<eBlocks, TB, 0, stream>>>(ei, E, ET, as1, ad1, m1, z1);
    tg_edge_scat1<<<e2Blocks, TB, 0, stream>>>(ei, E, ET, as1, ad1, m1, z1, h1, acc1);

    // layer 2 (ELU fused into GEMM A-load)
    tg_gemm2<<<mtiles, 64, 0, stream>>>(acc1, b1, W2, h2, N);
    tg_alpha2<<<(N + TB - 1) / TB, TB, 0, stream>>>(h2, a2_src, a2_dst, as2, ad2, N);
    tg_edge_max2<<<eBlocks, TB, 0, stream>>>(ei, E, ET, as2, ad2, m2);
    tg_edge_sum2<<<eBlocks, TB, 0, stream>>>(ei, E, ET, as2, ad2, m2, z2);
    tg_edge_scat2<<<eBlocks, TB, 0, stream>>>(ei, E, ET, as2, ad2, m2, z2, h2, out);
}